// VCRNet_24773371363530
// MI455X (gfx1250) — compile-verified
//
#include <hip/hip_runtime.h>
#include <math.h>

typedef float v2f __attribute__((ext_vector_type(2)));
typedef float v8f __attribute__((ext_vector_type(8)));

#define BATCH 256
#define DDIM  2048
#define CGRP  32
#define PDIM  64
#define PQ    4
#define NCOL  49
#define NPAD  64

// ---------------------------------------------------------------------------
// Fused GEMM + BN (+ optional residual), 2 batches per block:
//   Y[b,m,n] = s[m]*sum_k W[m,k]*X[b,k,n] + bias[m] (+R[b,m,n])
// W: [2048,2048] row-major, X: [B,2048,49], Y: [B,2048,49]
// block = 256 threads (8 waves); block computes 128 rows x 64(pad) cols x 2 batches.
// LDS X-panel stored K-paired so every B fragment is one ds_load_b64.
// ---------------------------------------------------------------------------
__global__ __launch_bounds__(256)
void gemm_bn_wmma(const float* __restrict__ W, const float* __restrict__ X,
                  const float* __restrict__ gamma, const float* __restrict__ beta,
                  const float* __restrict__ R, float* __restrict__ Y)
{
    // Bp[batch][kpair][col][2] : (K=2kp, K=2kp+1) adjacent -> single b64 frag load
    __shared__ float Bp[2][8][NPAD][2];
    const int b0     = blockIdx.z * 2;
    const int tid    = threadIdx.x;
    const int wave   = tid >> 5;
    const int lane   = tid & 31;
    const int lhalf  = lane >> 4;     // 0: lanes 0-15, 1: lanes 16-31
    const int l16    = lane & 15;
    const int rowBase = blockIdx.x * 128 + wave * 16;

    const float* Xb0 = X + (size_t)b0 * DDIM * NCOL;

    v8f acc[2][4];
    #pragma unroll
    for (int i = 0; i < 2; ++i)
        #pragma unroll
        for (int j = 0; j < 4; ++j) acc[i][j] = (v8f){};

    for (int k0 = 0; k0 < DDIM; k0 += 16) {
        __syncthreads();
        // stage 2 x 16 x 64 panel, K-paired; zero-pad cols 49..63
        for (int i = tid; i < 2 * 16 * NPAD; i += 256) {
            int bb  = i >> 10;
            int rem = i & 1023;
            int kk  = rem >> 6;
            int nn  = rem & 63;
            float v = 0.f;
            if (nn < NCOL) v = Xb0[(size_t)bb * DDIM * NCOL + (size_t)(k0 + kk) * NCOL + nn];
            Bp[bb][kk >> 1][nn][kk & 1] = v;
        }
        __syncthreads();

        const float* Wrow = W + (size_t)(rowBase + l16) * DDIM + k0;
        #pragma unroll
        for (int ks = 0; ks < 16; ks += 4) {
            const int kb = ks + 2 * lhalf;          // even
            const int kp = kb >> 1;
            v2f a = *(const v2f*)(Wrow + kb);       // 8B-aligned global b64
            #pragma unroll
            for (int bb = 0; bb < 2; ++bb) {
                v2f f0 = *(const v2f*)&Bp[bb][kp][ 0 + l16][0];
                v2f f1 = *(const v2f*)&Bp[bb][kp][16 + l16][0];
                v2f f2 = *(const v2f*)&Bp[bb][kp][32 + l16][0];
                v2f f3 = *(const v2f*)&Bp[bb][kp][48 + l16][0];
                acc[bb][0] = __builtin_amdgcn_wmma_f32_16x16x4_f32(false, a, false, f0, (short)0, acc[bb][0], false, false);
                acc[bb][1] = __builtin_amdgcn_wmma_f32_16x16x4_f32(false, a, false, f1, (short)0, acc[bb][1], false, false);
                acc[bb][2] = __builtin_amdgcn_wmma_f32_16x16x4_f32(false, a, false, f2, (short)0, acc[bb][2], false, false);
                acc[bb][3] = __builtin_amdgcn_wmma_f32_16x16x4_f32(false, a, false, f3, (short)0, acc[bb][3], false, false);
            }
        }
    }

    const float sc = rsqrtf(1.00001f);
    #pragma unroll
    for (int bb = 0; bb < 2; ++bb) {
        float* Yb = Y + (size_t)(b0 + bb) * DDIM * NCOL;
        const float* Rb = R ? (R + (size_t)(b0 + bb) * DDIM * NCOL) : nullptr;
        #pragma unroll
        for (int r = 0; r < 8; ++r) {
            int row = rowBase + r + 8 * lhalf;  // C/D layout: M=r / M=r+8 per lane half
            float s  = gamma[row] * sc;
            float bv = beta[row];
            float vals[4] = { acc[bb][0][r], acc[bb][1][r], acc[bb][2][r], acc[bb][3][r] };
            #pragma unroll
            for (int t4 = 0; t4 < 4; ++t4) {
                int col = t4 * 16 + l16;
                if (col < NCOL) {
                    float v = vals[t4] * s + bv;
                    if (Rb) v += Rb[(size_t)row * NCOL + col];
                    Yb[(size_t)row * NCOL + col] = v;
                }
            }
        }
    }
}

// ---------------------------------------------------------------------------
// Grouped 3x3 conv + BN via WMMA: per (b,g) GEMM of [64oc x 576k] x [576k x 49n]
// im2col panel built on the fly in LDS (K-paired); W2 rows are A fragments.
// block = 128 threads (4 waves, 16 oc rows each).
// ---------------------------------------------------------------------------
__global__ __launch_bounds__(128)
void conv3x3_bn_wmma(const float* __restrict__ Xc, const float* __restrict__ W2,
                     const float* __restrict__ g2, const float* __restrict__ b2,
                     float* __restrict__ Zc)
{
    __shared__ float Xt[64][50];
    __shared__ float Bp[8][NPAD][2];
    const int b = blockIdx.x >> 5;
    const int g = blockIdx.x & 31;
    const int tid   = threadIdx.x;
    const int wave  = tid >> 5;
    const int lane  = tid & 31;
    const int lhalf = lane >> 4;
    const int l16   = lane & 15;
    const int rowBase = wave * 16;   // oc rows

    const float* Xg = Xc + ((size_t)b * DDIM + g * 64) * NCOL;
    for (int i = tid; i < 64 * NCOL; i += 128) {
        int p = i / NCOL, n = i - p * NCOL;
        Xt[p][n] = Xg[(size_t)p * NCOL + n];
    }

    v8f acc0 = {}, acc1 = {}, acc2 = {}, acc3 = {};

    for (int k0 = 0; k0 < 576; k0 += 16) {
        __syncthreads();
        // build 16 x 64 im2col panel (K = ic*9 + ky*3 + kx), K-paired in LDS
        for (int i = tid; i < 16 * NPAD; i += 128) {
            int kk = i >> 6;
            int nn = i & 63;
            int k  = k0 + kk;
            int ic = k / 9;
            int r  = k - ic * 9;
            int ky = r / 3;
            int kx = r - ky * 3;
            float v = 0.f;
            if (nn < NCOL) {
                int h = nn / 7, w = nn - h * 7;
                int hh = h + ky - 1, ww = w + kx - 1;
                if (hh >= 0 && hh < 7 && ww >= 0 && ww < 7)
                    v = Xt[ic][hh * 7 + ww];
            }
            Bp[kk >> 1][nn][kk & 1] = v;
        }
        __syncthreads();

        const float* Wrow = W2 + (size_t)(g * 64 + rowBase + l16) * 576 + k0;
        #pragma unroll
        for (int ks = 0; ks < 16; ks += 4) {
            const int kb = ks + 2 * lhalf;
            const int kp = kb >> 1;
            v2f a = *(const v2f*)(Wrow + kb);
            v2f f0 = *(const v2f*)&Bp[kp][ 0 + l16][0];
            v2f f1 = *(const v2f*)&Bp[kp][16 + l16][0];
            v2f f2 = *(const v2f*)&Bp[kp][32 + l16][0];
            v2f f3 = *(const v2f*)&Bp[kp][48 + l16][0];
            acc0 = __builtin_amdgcn_wmma_f32_16x16x4_f32(false, a, false, f0, (short)0, acc0, false, false);
            acc1 = __builtin_amdgcn_wmma_f32_16x16x4_f32(false, a, false, f1, (short)0, acc1, false, false);
            acc2 = __builtin_amdgcn_wmma_f32_16x16x4_f32(false, a, false, f2, (short)0, acc2, false, false);
            acc3 = __builtin_amdgcn_wmma_f32_16x16x4_f32(false, a, false, f3, (short)0, acc3, false, false);
        }
    }

    const float sc = rsqrtf(1.00001f);
    #pragma unroll
    for (int r = 0; r < 8; ++r) {
        int row = rowBase + r + 8 * lhalf;
        int ch  = g * 64 + row;
        float s  = g2[ch] * sc;
        float bv = b2[ch];
        float vals[4] = { acc0[r], acc1[r], acc2[r], acc3[r] };
        #pragma unroll
        for (int t4 = 0; t4 < 4; ++t4) {
            int col = t4 * 16 + l16;
            if (col < NCOL)
                Zc[((size_t)b * DDIM + ch) * NCOL + col] = vals[t4] * s + bv;
        }
    }
}

// ---------------------------------------------------------------------------
// Per (b,c): pooled -> q -> k -> scores[b,c,n]
// ---------------------------------------------------------------------------
__global__ __launch_bounds__(64)
void qk_scores(const float* __restrict__ Zc,
               const float* __restrict__ Wq, const float* __restrict__ gq, const float* __restrict__ bq,
               const float* __restrict__ Wk, const float* __restrict__ gk, const float* __restrict__ bk,
               float* __restrict__ scores)
{
    __shared__ float Zt[64][50];
    __shared__ float pooled[64];
    __shared__ float qsh[PQ];
    int b = blockIdx.x >> 5;
    int c = blockIdx.x & 31;
    int tid = threadIdx.x;
    const float* Zg = Zc + ((size_t)b * DDIM + c * 64) * NCOL;
    for (int i = tid; i < 64 * NCOL; i += 64) {
        int p = i / NCOL, n = i - p * NCOL;
        Zt[p][n] = Zg[(size_t)p * NCOL + n];
    }
    __syncthreads();
    {
        float s = 0.f;
        for (int n = 0; n < NCOL; ++n) s += Zt[tid][n];
        pooled[tid] = s * (1.f / 49.f);
    }
    __syncthreads();
    const float sc = rsqrtf(1.00001f);
    if (tid < PQ) {
        const float* wq = Wq + ((size_t)c * PQ + tid) * 64;
        float s = 0.f;
        for (int p = 0; p < 64; ++p) s = fmaf(wq[p], pooled[p], s);
        qsh[tid] = gq[c * PQ + tid] * sc * s + bq[c * PQ + tid];
    }
    __syncthreads();
    if (tid < NCOL) {
        float accv = 0.f;
        #pragma unroll
        for (int pq = 0; pq < PQ; ++pq) {
            const float* wk = Wk + ((size_t)c * PQ + pq) * 64;
            float s = 0.f;
            for (int p = 0; p < 64; ++p) s = fmaf(wk[p], Zt[p][tid], s);
            float kv = gk[c * PQ + pq] * sc * s + bk[c * PQ + pq];
            accv = fmaf(qsh[pq], kv, accv);
        }
        scores[((size_t)b * CGRP + c) * NCOL + tid] = accv;
    }
}

// ---------------------------------------------------------------------------
// Softmax over BATCH axis (faithful torch Softmax2d quirk): one thread per (c,n)
// ---------------------------------------------------------------------------
__global__ __launch_bounds__(256)
void softmax_batch(const float* __restrict__ scores, float* __restrict__ M)
{
    int idx = blockIdx.x * 256 + threadIdx.x;
    if (idx >= CGRP * NCOL) return;
    const int stride = CGRP * NCOL;   // 1568
    float mx = -3.4e38f;
    for (int b = 0; b < BATCH; ++b) mx = fmaxf(mx, scores[(size_t)b * stride + idx]);
    float sum = 0.f;
    for (int b = 0; b < BATCH; ++b) {
        float e = expf(scores[(size_t)b * stride + idx] - mx);
        M[(size_t)b * stride + idx] = e;
        sum += e;
    }
    float inv = 1.f / sum;
    for (int b = 0; b < BATCH; ++b) M[(size_t)b * stride + idx] *= inv;
}

// ---------------------------------------------------------------------------
// Global max(M) -> single scalar (single block; deterministic, no atomics)
// ---------------------------------------------------------------------------
__global__ __launch_bounds__(256)
void max_reduce(const float* __restrict__ M, float* __restrict__ mx_out)
{
    __shared__ float red[256];
    float mx = 0.f;   // softmax outputs are positive
    for (size_t i = threadIdx.x; i < (size_t)BATCH * CGRP * NCOL; i += 256)
        mx = fmaxf(mx, M[i]);
    red[threadIdx.x] = mx;
    __syncthreads();
    for (int s = 128; s > 0; s >>= 1) {
        if ((int)threadIdx.x < s) red[threadIdx.x] = fmaxf(red[threadIdx.x], red[threadIdx.x + s]);
        __syncthreads();
    }
    if (threadIdx.x == 0) mx_out[0] = red[0];
}

// ---------------------------------------------------------------------------
// Per-b: zout -> hc -> A=tanh(BN(..)) -> mix=A^T@hc+hc -> hc_ = relu(BN(mix))
// ---------------------------------------------------------------------------
__global__ __launch_bounds__(256)
void chan_relate(const float* __restrict__ Zc, const float* __restrict__ M,
                 const float* __restrict__ Wout, const float* __restrict__ gout, const float* __restrict__ bout,
                 const float* __restrict__ Wcr, const float* __restrict__ gcr, const float* __restrict__ bcr,
                 const float* __restrict__ gcr1, const float* __restrict__ bcr1,
                 float* __restrict__ hc_out)
{
    __shared__ float Msh[CGRP * NCOL];
    __shared__ float zout[DDIM];
    __shared__ float hcsh[CGRP * PQ];
    __shared__ float Ash[CGRP * CGRP];
    int b = blockIdx.x;
    int tid = threadIdx.x;
    for (int i = tid; i < CGRP * NCOL; i += 256) Msh[i] = M[(size_t)b * CGRP * NCOL + i];
    __syncthreads();
    for (int j = 0; j < 8; ++j) {
        int ch = tid + j * 256;
        int c = ch >> 6;
        const float* zp = Zc + ((size_t)b * DDIM + ch) * NCOL;
        const float* mp = Msh + c * NCOL;
        float s = 0.f;
        for (int n = 0; n < NCOL; ++n) s = fmaf(zp[n], mp[n], s);
        zout[ch] = s;
    }
    __syncthreads();
    const float sc = rsqrtf(1.00001f);
    if (tid < CGRP * PQ) {
        int c = tid >> 2, pq = tid & 3;
        const float* wp = Wout + ((size_t)c * PQ + pq) * 64;
        const float* zp = zout + c * 64;
        float s = 0.f;
        for (int p = 0; p < 64; ++p) s = fmaf(wp[p], zp[p], s);
        hcsh[tid] = gout[tid] * sc * s + bout[tid];
    }
    __syncthreads();
    for (int t2 = tid; t2 < CGRP * CGRP; t2 += 256) {
        int cg = t2 >> 5, o = t2 & 31;
        const float* wp = Wcr + ((size_t)cg * CGRP + o) * PQ;
        float s = 0.f;
        #pragma unroll
        for (int pq = 0; pq < PQ; ++pq) s = fmaf(wp[pq], hcsh[cg * PQ + pq], s);
        Ash[t2] = tanhf(gcr[t2] * sc * s + bcr[t2]);   // A_orig[cg][o]
    }
    __syncthreads();
    if (tid < CGRP * PQ) {
        int i = tid >> 2, p = tid & 3;
        float m = hcsh[tid];
        for (int j = 0; j < CGRP; ++j)
            m = fmaf(Ash[j * CGRP + i], hcsh[j * PQ + p], m);   // swapaxes(1,2) applied
        float v = gcr1[tid] * sc * m + bcr1[tid];
        hc_out[(size_t)b * CGRP * PQ + tid] = fmaxf(v, 0.f);
    }
}

// ---------------------------------------------------------------------------
// Per-b: alpha/beta via factorization (Wa.hc_)·M_ + ba ; cm = Zc*relu(Zc*a+b)
// ---------------------------------------------------------------------------
__global__ __launch_bounds__(256)
void apply_mod(const float* __restrict__ Zc, const float* __restrict__ M,
               const float* __restrict__ hc, const float* __restrict__ mxp,
               const float* __restrict__ Wa, const float* __restrict__ ba,
               const float* __restrict__ Wb, const float* __restrict__ bb,
               float* __restrict__ cm)
{
    __shared__ float Msh[CGRP * NCOL];
    __shared__ float hcsh[CGRP * PQ];
    __shared__ float wa_h[DDIM];
    __shared__ float wb_h[DDIM];
    int b = blockIdx.x;
    int tid = threadIdx.x;
    float invmx = 1.f / mxp[0];
    for (int i = tid; i < CGRP * NCOL; i += 256) Msh[i] = M[(size_t)b * CGRP * NCOL + i] * invmx;
    if (tid < CGRP * PQ) hcsh[tid] = hc[(size_t)b * CGRP * PQ + tid];
    __syncthreads();
    for (int j = 0; j < 8; ++j) {
        int ch = tid + j * 256;
        int c = ch >> 6, p = ch & 63;
        const float* wap = Wa + ((size_t)c * PDIM + p) * PQ;
        const float* wbp = Wb + ((size_t)c * PDIM + p) * PQ;
        float sa = 0.f, sb = 0.f;
        #pragma unroll
        for (int pq = 0; pq < PQ; ++pq) {
            float h = hcsh[c * PQ + pq];
            sa = fmaf(wap[pq], h, sa);
            sb = fmaf(wbp[pq], h, sb);
        }
        wa_h[ch] = sa; wb_h[ch] = sb;
    }
    __syncthreads();
    for (int j = 0; j < 8; ++j) {
        int ch = tid + j * 256;
        int c = ch >> 6;
        const float* zp = Zc + ((size_t)b * DDIM + ch) * NCOL;
        float* cp = cm + ((size_t)b * DDIM + ch) * NCOL;
        float av = wa_h[ch], bv = wb_h[ch];
        float bav = ba[ch], bbv = bb[ch];
        const float* mp = Msh + c * NCOL;
        for (int n = 0; n < NCOL; ++n) {
            float z = zp[n];
            float al = fmaf(av, mp[n], bav);
            float be = fmaf(bv, mp[n], bbv);
            float x_ = fmaxf(fmaf(z, al, be), 0.f);
            cp[n] = z * x_;
        }
    }
}

// ---------------------------------------------------------------------------
extern "C" void kernel_launch(void* const* d_in, const int* in_sizes, int n_in,
                              void* d_out, int out_size, void* d_ws, size_t ws_size,
                              hipStream_t stream) {
    const float* x    = (const float*)d_in[0];
    const float* W1   = (const float*)d_in[1];
    const float* g1   = (const float*)d_in[2];
    const float* b1   = (const float*)d_in[3];
    const float* W2   = (const float*)d_in[4];
    const float* g2   = (const float*)d_in[5];
    const float* b2   = (const float*)d_in[6];
    const float* Wq   = (const float*)d_in[7];
    const float* gq   = (const float*)d_in[8];
    const float* bq   = (const float*)d_in[9];
    const float* Wk   = (const float*)d_in[10];
    const float* gk   = (const float*)d_in[11];
    const float* bk   = (const float*)d_in[12];
    const float* Wout = (const float*)d_in[13];
    const float* gout = (const float*)d_in[14];
    const float* bout = (const float*)d_in[15];
    const float* Wcr  = (const float*)d_in[16];
    const float* gcr  = (const float*)d_in[17];
    const float* bcr  = (const float*)d_in[18];
    const float* gcr1 = (const float*)d_in[19];
    const float* bcr1 = (const float*)d_in[20];
    const float* Wa   = (const float*)d_in[21];
    const float* ba   = (const float*)d_in[22];
    const float* Wb   = (const float*)d_in[23];
    const float* bb   = (const float*)d_in[24];
    const float* W3   = (const float*)d_in[25];
    const float* g3   = (const float*)d_in[26];
    const float* b3   = (const float*)d_in[27];

    float* ws = (float*)d_ws;
    const size_t TENS = (size_t)BATCH * DDIM * NCOL;    // 25,690,112
    float* Xc     = ws;                                  // reused as cm later
    float* Zc     = ws + TENS;
    float* scores = ws + 2 * TENS;
    float* Mbuf   = scores + (size_t)BATCH * CGRP * NCOL;
    float* hcbuf  = Mbuf + (size_t)BATCH * CGRP * NCOL;
    float* mxbuf  = hcbuf + (size_t)BATCH * CGRP * PQ;

    dim3 ggrid(DDIM / 128, 1, BATCH / 2);

    // 1) Xc = BN(W1 @ x)                                     [WMMA]
    gemm_bn_wmma<<<ggrid, 256, 0, stream>>>(W1, x, g1, b1, nullptr, Xc);
    // 2) Zc = BN(grouped 3x3 conv(Xc))                       [WMMA, im2col]
    conv3x3_bn_wmma<<<BATCH * CGRP, 128, 0, stream>>>(Xc, W2, g2, b2, Zc);
    // 3) scores[b,c,n]
    qk_scores<<<BATCH * CGRP, 64, 0, stream>>>(Zc, Wq, gq, bq, Wk, gk, bk, scores);
    // 4) M = softmax over batch axis
    softmax_batch<<<(CGRP * NCOL + 255) / 256, 256, 0, stream>>>(scores, Mbuf);
    // 5) max(M)
    max_reduce<<<1, 256, 0, stream>>>(Mbuf, mxbuf);
    // 6) hc_ per batch
    chan_relate<<<BATCH, 256, 0, stream>>>(Zc, Mbuf, Wout, gout, bout,
                                           Wcr, gcr, bcr, gcr1, bcr1, hcbuf);
    // 7) cm (reuses Xc buffer)
    apply_mod<<<BATCH, 256, 0, stream>>>(Zc, Mbuf, hcbuf, mxbuf, Wa, ba, Wb, bb, Xc);
    // 8) OUT = BN(W3 @ cm) + x                               [WMMA]
    gemm_bn_wmma<<<ggrid, 256, 0, stream>>>(W3, Xc, g3, b3, x, (float*)d_out);
}